// GraphRespiratory_75788992905512
// MI455X (gfx1250) — compile-verified
//
#include <hip/hip_runtime.h>
#include <hip/hip_bf16.h>

#define N_NODES 50000
#define N_EDGES 800000
#define EP_EDGES (N_EDGES + N_NODES)   // with self loops
#define IN_CH 128
#define HID 64
#define HEADS 2

typedef float v2f __attribute__((ext_vector_type(2)));
typedef float v8f __attribute__((ext_vector_type(8)));

// ---------- helpers ----------
__device__ __forceinline__ unsigned encF(float f) {
    unsigned u = __float_as_uint(f);
    return (u >> 31) ? ~u : (u | 0x80000000u);
}
__device__ __forceinline__ float decF(unsigned u) {
    return __uint_as_float((u >> 31) ? (u & 0x7FFFFFFFu) : ~u);
}

__global__ void zero_kernel(float* __restrict__ p, long n) {
    long i = (long)blockIdx.x * blockDim.x + threadIdx.x;
    if (i < n) p[i] = 0.0f;
}

// ---------- self-loop edge attr: segment mean of edge_attr over dst ----------
__global__ void loop_sum_kernel(const int* __restrict__ ei, const float* __restrict__ eattr,
                                float* __restrict__ sumA, float* __restrict__ cnt) {
    int e = blockIdx.x * blockDim.x + threadIdx.x;
    if (e >= N_EDGES) return;
    int d = ei[N_EDGES + e];
    atomicAdd(&cnt[d], 1.0f);
    const float* ap = eattr + (size_t)e * 16;
    float* sp = sumA + (size_t)d * 16;
#pragma unroll
    for (int k = 0; k < 16; ++k) atomicAdd(&sp[k], ap[k]);
}

__global__ void loop_div_kernel(float* __restrict__ sumA, const float* __restrict__ cnt) {
    int t = blockIdx.x * blockDim.x + threadIdx.x;
    if (t >= N_NODES * 16) return;
    float c = cnt[t >> 4];
    sumA[t] /= (c > 1.0f ? c : 1.0f);
}

// ---------- collapse We (16x128) against att_edge (2x64) -> wc[16][2] ----------
__global__ void wecomb_kernel(const float* __restrict__ We, const float* __restrict__ ae,
                              float* __restrict__ wc) {
    int t = threadIdx.x;
    if (t >= 32) return;
    int d = t >> 1, hh = t & 1;
    float s = 0.0f;
#pragma unroll 8
    for (int c = 0; c < 64; ++c) s += We[d * 128 + hh * 64 + c] * ae[hh * 64 + c];
    wc[d * 2 + hh] = s;
}

// ---------- fp32 WMMA GEMM: C[M x 128] = A[M x K] * B[K x 128], M % 16 == 0 ----------
// block = 256 threads (8 waves); wave w owns 16x16 output tile (rows blockIdx.x*16, cols w*16)
template <int K>
__global__ void __launch_bounds__(256) wmma_gemm_n128(const float* __restrict__ A,
                                                      const float* __restrict__ B,
                                                      float* __restrict__ C) {
    const int lane = threadIdx.x & 31;
    const int wave = threadIdx.x >> 5;
    const int half = lane >> 4;   // 0: K pair {k,k+1}, 1: {k+2,k+3} (A frag) / rows +8 (C frag)
    const int l16  = lane & 15;
    const int row0 = blockIdx.x << 4;
    const int col0 = wave << 4;

    const float* Ar = A + (size_t)(row0 + l16) * K + 2 * half;  // 8B aligned
    const float* B0 = B + col0 + l16;

    v8f acc = {};
#pragma unroll
    for (int k = 0; k < K; k += 4) {
        v2f a = *(const v2f*)(Ar + k);                 // A[m][k+2h], A[m][k+2h+1]
        v2f b;
        b.x = B0[(size_t)(k + 2 * half)     * 128];    // B[k+2h][n]
        b.y = B0[(size_t)(k + 2 * half + 1) * 128];    // B[k+2h+1][n]
        acc = __builtin_amdgcn_wmma_f32_16x16x4_f32(false, a, false, b,
                                                    (short)0, acc, false, false);
    }
    float* Cp = C + (size_t)(row0 + 8 * half) * 128 + col0 + l16;
#pragma unroll
    for (int i = 0; i < 8; ++i) Cp[(size_t)i * 128] = acc[i];
}

// ---------- per-node attention scalars: asrc/adst [N,2] ----------
__global__ void att_scalar_kernel(const float* __restrict__ hf, const float* __restrict__ as_,
                                  const float* __restrict__ ad_, float* __restrict__ asrc,
                                  float* __restrict__ adst) {
    int t = blockIdx.x * blockDim.x + threadIdx.x;
    if (t >= N_NODES * 2) return;
    int n = t >> 1, hh = t & 1;
    const float* hp = hf + (size_t)n * 128 + hh * 64;
    float s1 = 0.0f, s2 = 0.0f;
#pragma unroll 8
    for (int c = 0; c < 64; ++c) {
        float v = hp[c];
        s1 += v * as_[hh * 64 + c];
        s2 += v * ad_[hh * 64 + c];
    }
    asrc[t] = s1;
    adst[t] = s2;
}

// ---------- per-edge alpha (leaky relu) + segment max via ordered-uint atomicMax ----------
__global__ void alpha_max_kernel(const int* __restrict__ ei, const float* __restrict__ eattr,
                                 const float* __restrict__ loopA, const float* __restrict__ wc,
                                 const float* __restrict__ asrc, const float* __restrict__ adst,
                                 float* __restrict__ alpha, unsigned* __restrict__ amaxU) {
    int e = blockIdx.x * blockDim.x + threadIdx.x;
    if (e >= EP_EDGES) return;
    int s, d;
    const float* ap;
    if (e < N_EDGES) { s = ei[e]; d = ei[N_EDGES + e]; ap = eattr + (size_t)e * 16; }
    else             { s = d = e - N_EDGES;            ap = loopA + (size_t)(e - N_EDGES) * 16; }
    float ae0 = 0.0f, ae1 = 0.0f;
#pragma unroll
    for (int k = 0; k < 16; ++k) {
        float a = ap[k];
        ae0 += a * wc[k * 2 + 0];
        ae1 += a * wc[k * 2 + 1];
    }
#pragma unroll
    for (int hh = 0; hh < 2; ++hh) {
        float al = asrc[s * 2 + hh] + adst[d * 2 + hh] + (hh ? ae1 : ae0);
        al = (al > 0.0f) ? al : 0.2f * al;
        alpha[(size_t)e * 2 + hh] = al;
        atomicMax(&amaxU[d * 2 + hh], encF(al));
    }
}

// ---------- exp(alpha - amax[dst]) + segment sum; alpha overwritten with ex ----------
__global__ void exp_sum_kernel(const int* __restrict__ ei, float* __restrict__ alpha,
                               const unsigned* __restrict__ amaxU, float* __restrict__ denom) {
    int e = blockIdx.x * blockDim.x + threadIdx.x;
    if (e >= EP_EDGES) return;
    int d = (e < N_EDGES) ? ei[N_EDGES + e] : (e - N_EDGES);
#pragma unroll
    for (int hh = 0; hh < 2; ++hh) {
        float m  = decF(amaxU[d * 2 + hh]);
        float ex = __expf(alpha[(size_t)e * 2 + hh] - m);
        alpha[(size_t)e * 2 + hh] = ex;
        atomicAdd(&denom[d * 2 + hh], ex);
    }
}

// ---------- message + scatter aggregate: one wave32 per edge, float4 per lane ----------
__global__ void aggregate_kernel(const int* __restrict__ ei, const float* __restrict__ ex,
                                 const float* __restrict__ denom, const float* __restrict__ hf,
                                 float* __restrict__ acc) {
    long gt = (long)blockIdx.x * blockDim.x + threadIdx.x;
    long e = gt >> 5;
    if (e >= EP_EDGES) return;
    int lane = (int)(gt & 31);
    int s, d;
    if (e < N_EDGES) { s = ei[e]; d = ei[N_EDGES + e]; } else { s = d = (int)(e - N_EDGES); }
    int head = lane >> 4;  // features [0,64) head 0, [64,128) head 1
    float attn = ex[e * 2 + head] / (denom[d * 2 + head] + 1e-16f);
    const float4 hv = *(const float4*)(hf + (size_t)s * 128 + lane * 4);
    float* apx = acc + (size_t)d * 128 + lane * 4;
    atomicAdd(apx + 0, attn * hv.x);
    atomicAdd(apx + 1, attn * hv.y);
    atomicAdd(apx + 2, attn * hv.z);
    atomicAdd(apx + 3, attn * hv.w);
}

// ---------- mean over heads + bias (+ optional relu) ----------
__global__ void finalize_kernel(const float* __restrict__ acc, const float* __restrict__ bias,
                                float* __restrict__ out, int do_relu) {
    int t = blockIdx.x * blockDim.x + threadIdx.x;
    if (t >= N_NODES * 64) return;
    int n = t >> 6, c = t & 63;
    float v = 0.5f * (acc[(size_t)n * 128 + c] + acc[(size_t)n * 128 + 64 + c]) + bias[c];
    if (do_relu) v = fmaxf(v, 0.0f);
    out[t] = v;
}

// ---------- workspace layout (float offsets, float4-aligned where needed) ----------
#define OFF_LOOP  0L                       // N*16      = 800000
#define OFF_CNT   800000L                  // N         = 50000
#define OFF_WC    850048L                  // 32
#define OFF_H     850112L                  // N*128     = 6400000
#define OFF_ASRC  7250112L                 // N*2
#define OFF_ADST  7350112L                 // N*2
#define OFF_ALPHA 7450112L                 // EP*2      = 1700000
#define OFF_AMAX  9150112L                 // N*2 (as uint)
#define OFF_DEN   9250112L                 // N*2
#define OFF_ACC   9350112L                 // N*128     = 6400000
#define OFF_H1    15750112L                // N*64      = 3200000

static inline int cdiv(long a, long b) { return (int)((a + b - 1) / b); }

extern "C" void kernel_launch(void* const* d_in, const int* in_sizes, int n_in,
                              void* d_out, int out_size, void* d_ws, size_t ws_size,
                              hipStream_t stream) {
    const float* x     = (const float*)d_in[0];
    const int*   ei    = (const int*)d_in[1];     // [2,E]: rows src, dst
    const float* eattr = (const float*)d_in[2];
    const float* W1    = (const float*)d_in[3];
    const float* We1   = (const float*)d_in[4];
    const float* as1   = (const float*)d_in[5];
    const float* ad1   = (const float*)d_in[6];
    const float* ae1   = (const float*)d_in[7];
    const float* b1    = (const float*)d_in[8];
    const float* W2    = (const float*)d_in[9];
    const float* We2   = (const float*)d_in[10];
    const float* as2   = (const float*)d_in[11];
    const float* ad2   = (const float*)d_in[12];
    const float* ae2   = (const float*)d_in[13];
    const float* b2    = (const float*)d_in[14];
    float* out = (float*)d_out;

    float* ws = (float*)d_ws;
    float*    loopA = ws + OFF_LOOP;
    float*    cnt   = ws + OFF_CNT;
    float*    wc    = ws + OFF_WC;
    float*    hbuf  = ws + OFF_H;
    float*    asrc  = ws + OFF_ASRC;
    float*    adst  = ws + OFF_ADST;
    float*    alpha = ws + OFF_ALPHA;
    unsigned* amaxU = (unsigned*)(ws + OFF_AMAX);
    float*    den   = ws + OFF_DEN;
    float*    acc   = ws + OFF_ACC;
    float*    h1out = ws + OFF_H1;

    const int T = 256;

    // ----- self-loop edge attr (segment mean over dst) -----
    zero_kernel<<<cdiv(N_NODES * 16, T), T, 0, stream>>>(loopA, (long)N_NODES * 16);
    zero_kernel<<<cdiv(N_NODES, T), T, 0, stream>>>(cnt, (long)N_NODES);
    loop_sum_kernel<<<cdiv(N_EDGES, T), T, 0, stream>>>(ei, eattr, loopA, cnt);
    loop_div_kernel<<<cdiv(N_NODES * 16, T), T, 0, stream>>>(loopA, cnt);

    // ================= layer 1 =================
    wecomb_kernel<<<1, 32, 0, stream>>>(We1, ae1, wc);
    wmma_gemm_n128<128><<<N_NODES / 16, 256, 0, stream>>>(x, W1, hbuf);
    att_scalar_kernel<<<cdiv(N_NODES * 2, T), T, 0, stream>>>(hbuf, as1, ad1, asrc, adst);
    zero_kernel<<<cdiv(N_NODES * 2, T), T, 0, stream>>>((float*)amaxU, (long)N_NODES * 2);
    zero_kernel<<<cdiv(N_NODES * 2, T), T, 0, stream>>>(den, (long)N_NODES * 2);
    zero_kernel<<<cdiv((long)N_NODES * 128, T), T, 0, stream>>>(acc, (long)N_NODES * 128);
    alpha_max_kernel<<<cdiv(EP_EDGES, T), T, 0, stream>>>(ei, eattr, loopA, wc, asrc, adst, alpha, amaxU);
    exp_sum_kernel<<<cdiv(EP_EDGES, T), T, 0, stream>>>(ei, alpha, amaxU, den);
    aggregate_kernel<<<cdiv((long)EP_EDGES * 32, T), T, 0, stream>>>(ei, alpha, den, hbuf, acc);
    finalize_kernel<<<cdiv(N_NODES * 64, T), T, 0, stream>>>(acc, b1, h1out, 1);

    // ================= layer 2 =================
    wecomb_kernel<<<1, 32, 0, stream>>>(We2, ae2, wc);
    wmma_gemm_n128<64><<<N_NODES / 16, 256, 0, stream>>>(h1out, W2, hbuf);
    att_scalar_kernel<<<cdiv(N_NODES * 2, T), T, 0, stream>>>(hbuf, as2, ad2, asrc, adst);
    zero_kernel<<<cdiv(N_NODES * 2, T), T, 0, stream>>>((float*)amaxU, (long)N_NODES * 2);
    zero_kernel<<<cdiv(N_NODES * 2, T), T, 0, stream>>>(den, (long)N_NODES * 2);
    zero_kernel<<<cdiv((long)N_NODES * 128, T), T, 0, stream>>>(acc, (long)N_NODES * 128);
    alpha_max_kernel<<<cdiv(EP_EDGES, T), T, 0, stream>>>(ei, eattr, loopA, wc, asrc, adst, alpha, amaxU);
    exp_sum_kernel<<<cdiv(EP_EDGES, T), T, 0, stream>>>(ei, alpha, amaxU, den);
    aggregate_kernel<<<cdiv((long)EP_EDGES * 32, T), T, 0, stream>>>(ei, alpha, den, hbuf, acc);
    finalize_kernel<<<cdiv(N_NODES * 64, T), T, 0, stream>>>(acc, b2, out, 0);
}